// NestedGNN_32409823216461
// MI455X (gfx1250) — compile-verified
//
#include <hip/hip_runtime.h>

typedef __attribute__((ext_vector_type(2))) float v2f;
typedef __attribute__((ext_vector_type(8))) float v8f;

#define D 128
#define TILE_ROWS 64
#define LDS_PITCH 132   // 128 + 4 pad: conflict-free column-strided LDS reads

// ---------------------------------------------------------------- zero
__global__ void zero_kernel(float* __restrict__ p, size_t n) {
  size_t i = (size_t)blockIdx.x * blockDim.x + threadIdx.x;
  if (i < n) p[i] = 0.0f;
}

// ---------------------------------------------------------------- edge scatter
// 32 threads per edge, float4 per thread, native f32 atomics (resolve in L2)
__global__ void scatter_kernel(const float* __restrict__ h,
                               const int* __restrict__ ei,
                               float* __restrict__ agg, int E) {
  size_t gid = (size_t)blockIdx.x * blockDim.x + threadIdx.x;
  size_t e = gid >> 5;
  if (e >= (size_t)E) return;
  int c4 = ((int)gid & 31) << 2;
  int src = ei[e];
  int dst = ei[(size_t)E + e];
  const float4 v = *(const float4*)(h + (size_t)src * D + c4);
  float* a = agg + (size_t)dst * D + c4;
  unsafeAtomicAdd(a + 0, v.x);
  unsafeAtomicAdd(a + 1, v.y);
  unsafeAtomicAdd(a + 2, v.z);
  unsafeAtomicAdd(a + 3, v.w);
}

// ---------------------------------------------------------------- fused GIN MLP
// out = relu(((1+eps)*h + agg) @ W1 + b1) @ W2 + b2, fp32 WMMA 16x16x4
// 8 independent accumulators (one per 16-col tile) -> 8-way XDL ILP
__global__ __launch_bounds__(128) void gin_mlp_kernel(
    const float* __restrict__ h, const float* __restrict__ agg,
    const float* __restrict__ epsArr, int layer,
    const float* __restrict__ W1, const float* __restrict__ b1v,
    const float* __restrict__ W2, const float* __restrict__ b2v,
    float* __restrict__ out, int nrows) {
  extern __shared__ float smem[];
  float* sZ = smem;                            // 64 x 132
  float* sY = smem + TILE_ROWS * LDS_PITCH;    // 64 x 132

  const int tid  = threadIdx.x;
  const int wave = tid >> 5;
  const int lane = tid & 31;
  const int kh   = lane >> 4;    // K half-select of fragment
  const int nl   = lane & 15;    // M/N sub-index

  const int rowBase = blockIdx.x * TILE_ROWS;
  const float eps1 = 1.0f + epsArr[layer];

  // warm caches for the block-invariant weights
  __builtin_prefetch(W1 + (size_t)tid * D, 0, 0);
  __builtin_prefetch(W2 + (size_t)tid * D, 0, 0);

  // stage z = (1+eps)*h + agg into LDS (zeros for out-of-range rows)
  for (int i = tid; i < TILE_ROWS * (D / 4); i += 128) {
    int r  = i >> 5;
    int c4 = (i & 31) << 2;
    int row = rowBase + r;
    float4 v = make_float4(0.f, 0.f, 0.f, 0.f);
    if (row < nrows) {
      float4 hv = *(const float4*)(h   + (size_t)row * D + c4);
      float4 av = *(const float4*)(agg + (size_t)row * D + c4);
      v.x = eps1 * hv.x + av.x;
      v.y = eps1 * hv.y + av.y;
      v.z = eps1 * hv.z + av.z;
      v.w = eps1 * hv.w + av.w;
    }
    *(float4*)(sZ + r * LDS_PITCH + c4) = v;
  }
  __syncthreads();

  const int lr = wave * 16;      // this wave's private 16-row tile in LDS
  const int am = lr + nl;        // LDS row feeding the A fragment

  // ---------------- GEMM1: Y = relu(Z @ W1 + b1) ----------------
  {
    v8f acc[8];
#pragma unroll
    for (int t = 0; t < 8; ++t) acc[t] = (v8f){};

#pragma unroll 2
    for (int s = 0; s < 32; ++s) {
      const v2f af = *(const v2f*)(sZ + am * LDS_PITCH + 4 * s + 2 * kh);
      const float* wr = W1 + (size_t)(4 * s + 2 * kh) * D + nl;
#pragma unroll
      for (int t = 0; t < 8; ++t) {
        v2f bf; bf.x = wr[t * 16]; bf.y = wr[D + t * 16];
        acc[t] = __builtin_amdgcn_wmma_f32_16x16x4_f32(false, af, false, bf,
                                                       (short)0, acc[t],
                                                       false, false);
      }
    }

#pragma unroll
    for (int t = 0; t < 8; ++t) {
      const int ncol = t * 16;
      const float bias = b1v[ncol + nl];
#pragma unroll
      for (int v = 0; v < 8; ++v) {
        float y = fmaxf(acc[t][v] + bias, 0.0f);
        int m = v + 8 * kh;
        sY[(lr + m) * LDS_PITCH + ncol + nl] = y;  // wave-private tile
      }
    }
  }

  // ---------------- GEMM2: H' = Y @ W2 + b2 ----------------
  {
    v8f acc[8];
#pragma unroll
    for (int t = 0; t < 8; ++t) acc[t] = (v8f){};

#pragma unroll 2
    for (int s = 0; s < 32; ++s) {
      const v2f af = *(const v2f*)(sY + am * LDS_PITCH + 4 * s + 2 * kh);
      const float* wr = W2 + (size_t)(4 * s + 2 * kh) * D + nl;
#pragma unroll
      for (int t = 0; t < 8; ++t) {
        v2f bf; bf.x = wr[t * 16]; bf.y = wr[D + t * 16];
        acc[t] = __builtin_amdgcn_wmma_f32_16x16x4_f32(false, af, false, bf,
                                                       (short)0, acc[t],
                                                       false, false);
      }
    }

#pragma unroll
    for (int t = 0; t < 8; ++t) {
      const int ncol = t * 16;
      const float bias = b2v[ncol + nl];
#pragma unroll
      for (int v = 0; v < 8; ++v) {
        int m = v + 8 * kh;
        int grow = rowBase + lr + m;
        if (grow < nrows)
          out[(size_t)grow * D + ncol + nl] = acc[t][v] + bias;
      }
    }
  }
}

// ---------------------------------------------------------------- batch pooling
__global__ void pool_kernel(const float* __restrict__ h1,
                            const float* __restrict__ h2,
                            const float* __restrict__ h3,
                            const int* __restrict__ batch,
                            float* __restrict__ sums,
                            float* __restrict__ cnt, int n) {
  size_t gid = (size_t)blockIdx.x * blockDim.x + threadIdx.x;
  if (gid >= (size_t)n * 384) return;
  int node = (int)(gid / 384);
  int c = (int)(gid % 384);
  int b = batch[node];
  int layer = c >> 7;
  int cc = c & 127;
  const float* hl = (layer == 0) ? h1 : (layer == 1 ? h2 : h3);
  unsafeAtomicAdd(&sums[(size_t)b * 384 + c], hl[(size_t)node * D + cc]);
  if (c == 0) unsafeAtomicAdd(&cnt[b], 1.0f);
}

// ---------------------------------------------------------------- head MLP (tiny)
__global__ __launch_bounds__(384) void head_kernel(
    const float* __restrict__ sums, const float* __restrict__ cnt,
    const float* __restrict__ l1w, const float* __restrict__ l1b,
    const float* __restrict__ l2w, const float* __restrict__ l2b,
    float* __restrict__ out) {
  __shared__ float g[384];
  __shared__ float g1[128];
  int tid = threadIdx.x;
  float s = 0.0f;
  for (int b = 0; b < 64; ++b)
    s += sums[b * 384 + tid] / fmaxf(cnt[b], 1.0f);
  g[tid] = s;
  __syncthreads();
  if (tid < 128) {
    float a = l1b[tid];
    for (int c = 0; c < 384; ++c) a += g[c] * l1w[c * 128 + tid];
    g1[tid] = fmaxf(a, 0.0f);
  }
  __syncthreads();
  if (tid < 64) {
    float a = l2b[tid];
    for (int j = 0; j < 128; ++j) a += g1[j] * l2w[j * 64 + tid];
    out[tid] = fmaxf(a, 0.0f);
  }
}

// ---------------------------------------------------------------- launch
extern "C" void kernel_launch(void* const* d_in, const int* in_sizes, int n_in,
                              void* d_out, int out_size, void* d_ws, size_t ws_size,
                              hipStream_t stream) {
  const float* x     = (const float*)d_in[0];
  const int*   ei    = (const int*)d_in[1];
  const int*   batch = (const int*)d_in[2];
  const float* eps   = (const float*)d_in[3];
  const float* W1    = (const float*)d_in[4];
  const float* b1    = (const float*)d_in[5];
  const float* W2    = (const float*)d_in[6];
  const float* b2    = (const float*)d_in[7];
  const float* l1w   = (const float*)d_in[8];
  const float* l1b   = (const float*)d_in[9];
  const float* l2w   = (const float*)d_in[10];
  const float* l2b   = (const float*)d_in[11];
  float* out = (float*)d_out;

  const int N = in_sizes[0] / D;
  const int E = in_sizes[1] / 2;
  const size_t NP = (size_t)N * D;

  float* ws = (float*)d_ws;
  float* hbuf[3] = { ws, ws + NP, ws + 2 * NP };
  float* agg  = ws + 3 * NP;
  float* sums = ws + 4 * NP;
  float* cnt  = sums + 64 * 384;

  const int mlpBlocks = (N + TILE_ROWS - 1) / TILE_ROWS;
  const size_t lds = (size_t)2 * TILE_ROWS * LDS_PITCH * sizeof(float);

  const float* hin = x;
  for (int l = 0; l < 3; ++l) {
    zero_kernel<<<(unsigned)((NP + 255) / 256), 256, 0, stream>>>(agg, NP);
    size_t sth = (size_t)E * 32;
    scatter_kernel<<<(unsigned)((sth + 255) / 256), 256, 0, stream>>>(hin, ei, agg, E);
    gin_mlp_kernel<<<mlpBlocks, 128, lds, stream>>>(
        hin, agg, eps, l,
        W1 + (size_t)l * D * D, b1 + (size_t)l * D,
        W2 + (size_t)l * D * D, b2 + (size_t)l * D,
        hbuf[l], N);
    hin = hbuf[l];
  }

  size_t psz = 64 * 384 + 64;
  zero_kernel<<<(unsigned)((psz + 255) / 256), 256, 0, stream>>>(sums, psz);
  size_t pth = (size_t)N * 384;
  pool_kernel<<<(unsigned)((pth + 255) / 256), 256, 0, stream>>>(
      hbuf[0], hbuf[1], hbuf[2], batch, sums, cnt, N);
  head_kernel<<<1, 384, 0, stream>>>(sums, cnt, l1w, l1b, l2w, l2b, out);
}